// MVBTSNet_13477607375191
// MI455X (gfx1250) — compile-verified
//
#include <hip/hip_runtime.h>

typedef __attribute__((ext_vector_type(2))) float v2f;
typedef __attribute__((ext_vector_type(8))) float v8f;

#define N_B   4
#define NV    4
#define CCH   64
#define HH    96
#define WW    320
#define NPTS  50000
#define EPSV  1e-3f
#define FEAT_DIM 103                 /* 64 sampled + 3 + 36 posenc */
#define WAVES 4                      /* 128-thread block  */
#define PTSW  16                     /* points per wave   */
#define STAGE_F (PTSW * NV * FEAT_DIM) /* 6592 floats = 26,368 B per wave */

__global__ __launch_bounds__(128)
void mvbts_kernel(const float* __restrict__ xyz,
                  const float* __restrict__ fm,
                  const float* __restrict__ poses,
                  const float* __restrict__ Ks,
                  const float* __restrict__ emptyf,
                  float* __restrict__ outFeat,
                  float* __restrict__ outInv) {
  __shared__ float Mlds[NV][12];              // fused 3x4 projection, [v][i*4+k]
  __shared__ float projLds[WAVES][16][16];    // WMMA D scatter buffer
  __shared__ float stage[WAVES][STAGE_F];     // per-wave output staging (coalesced flush)
  __shared__ float stageInv[WAVES][PTSW * NV];

  const int n = blockIdx.y;
  const int t = threadIdx.x;

  // ---- Phase 0: M[i][k] = sum_j K[i][j]*P[j][k]  (one entry per thread, 48 total)
  if (t < NV * 12) {
    const int v = t / 12, r = t % 12, i = r >> 2, k = r & 3;
    const float* Kp = Ks    + (size_t)(n * NV + v) * 9;
    const float* Pp = poses + (size_t)(n * NV + v) * 16;
    float acc = 0.f;
#pragma unroll
    for (int j = 0; j < 3; ++j) acc = fmaf(Kp[i * 3 + j], Pp[j * 4 + k], acc);
    Mlds[v][r] = acc;
  }
  __syncthreads();

  const int wave = t >> 5;
  const int lane = t & 31;
  const int p0   = blockIdx.x * (WAVES * PTSW) + wave * PTSW;  // NPTS % 16 == 0

  if (p0 < NPTS) {
    // ---- Phase 1: projection via V_WMMA_F32_16X16X4_F32 (16 pts x 4 views per op)
    const int ptl = lane & 15;
    const float* xp = xyz + ((size_t)n * NPTS + (size_t)(p0 + ptl)) * 3;
    const float px = xp[0], py = xp[1], pz = xp[2];
    v2f a, b;
    a.x = (lane < 16) ? px : pz;     // A 16x4: lanes 0-15 K=0,1 ; lanes 16-31 K=2,3
    a.y = (lane < 16) ? py : 1.0f;
    {
      const int col = lane & 15;     // B 4x16: col = v*3+i, B[k][col] = M_v[i][k]
      const int kb  = (lane >> 4) * 2;
      if (col < 12) {
        const int v = col / 3, i = col % 3;
        b.x = Mlds[v][i * 4 + kb];
        b.y = Mlds[v][i * 4 + kb + 1];
      } else {
        b.x = 0.f; b.y = 0.f;
      }
    }
    v8f c = {0.f, 0.f, 0.f, 0.f, 0.f, 0.f, 0.f, 0.f};
    v8f d = __builtin_amdgcn_wmma_f32_16x16x4_f32(
        false, a, false, b, (short)0, c, false, false);
#pragma unroll
    for (int j = 0; j < 8; ++j)      // D: VGPR j -> row j+8*(lane>=16), col lane&15
      projLds[wave][j + 8 * (lane >> 4)][lane & 15] = d[j];
  }
  __syncthreads();

  if (p0 < NPTS) {
    // ---- Phase 2: finish 2 point-views per lane, write into LDS staging
    const size_t fmn = (size_t)n * NV * CCH * HH * WW;
#pragma unroll
    for (int vv = 0; vv < 2; ++vv) {
      const int ptl = lane & 15;
      const int v   = (lane >> 4) + 2 * vv;

      const float pr0 = projLds[wave][ptl][v * 3 + 0];
      const float pr1 = projLds[wave][ptl][v * 3 + 1];
      const float z   = projLds[wave][ptl][v * 3 + 2];

      const float zm    = fmaxf(z, EPSV);
      const float inv_z = __builtin_amdgcn_rcpf(zm);   // v_rcp_f32, ~1 ULP
      const float xy0   = pr0 * inv_z;
      const float xy1   = pr1 * inv_z;
      const bool invalid = (z <= EPSV) || (xy0 < -1.f) || (xy0 > 1.f) ||
                           (xy1 < -1.f) || (xy1 > 1.f);
      float zc = (inv_z - (1.f / 80.f)) * (1.f / (1.f / 3.f - 1.f / 80.f));
      zc = 2.f * zc - 1.f;

      float* op = &stage[wave][(ptl * NV + v) * FEAT_DIM];

      // bilinear sample w/ border clamp
      const float gx = fminf(fmaxf(((xy0 + 1.f) * WW - 1.f) * 0.5f, 0.f), (float)(WW - 1));
      const float gy = fminf(fmaxf(((xy1 + 1.f) * HH - 1.f) * 0.5f, 0.f), (float)(HH - 1));
      const float x0 = floorf(gx), y0 = floorf(gy);
      const float wx = gx - x0,   wy = gy - y0;
      const int x0i = (int)x0, y0i = (int)y0;
      const int x1i = min(x0i + 1, WW - 1);
      const int y1i = min(y0i + 1, HH - 1);
      const float w00 = (1.f - wx) * (1.f - wy), w01 = wx * (1.f - wy);
      const float w10 = (1.f - wx) * wy,         w11 = wx * wy;
      const int o00 = y0i * WW + x0i, o01 = y0i * WW + x1i;
      const int o10 = y1i * WW + x0i, o11 = y1i * WW + x1i;
      const float* fb = fm + fmn + (size_t)v * CCH * HH * WW;

      if (invalid) {
#pragma unroll 8
        for (int cch = 0; cch < CCH; ++cch) op[cch] = emptyf[cch];
      } else {
#pragma unroll 8
        for (int cch = 0; cch < CCH; ++cch) {
          const float* f0 = fb + (size_t)cch * (HH * WW);
          op[cch] = w00 * f0[o00] + w01 * f0[o01] + w10 * f0[o10] + w11 * f0[o11];
        }
      }

      // posenc: [xy0, xy1, zc, {sin,cos}(pi*2^f * .) interleaved per reference]
      op[64] = xy0; op[65] = xy1; op[66] = zc;
#pragma unroll
      for (int f = 0; f < 6; ++f) {
        const float fr = 3.14159265358979323846f * (float)(1 << f);
        const float ax = fr * xy0, ay = fr * xy1, az = fr * zc;
        op[67 + f * 6 + 0] = __sinf(ax);
        op[67 + f * 6 + 1] = __sinf(ay);
        op[67 + f * 6 + 2] = __sinf(az);
        op[67 + f * 6 + 3] = __cosf(ax);
        op[67 + f * 6 + 4] = __cosf(ay);
        op[67 + f * 6 + 5] = __cosf(az);
      }
      stageInv[wave][ptl * NV + v] = invalid ? 1.f : 0.f;
    }

    // ---- Phase 3: coalesced flush. Wave output span is contiguous & 16B-aligned:
    // start elem = (n*NPTS + p0)*NV*FEAT_DIM, p0 % 16 == 0 -> multiple of 4.
    const float4* sp = (const float4*)&stage[wave][0];
    float4* gp = (float4*)(outFeat + ((size_t)n * NPTS + p0) * NV * FEAT_DIM);
    for (int i = lane; i < STAGE_F / 4; i += 32)   // 52 iters of b128, coalesced
      gp[i] = sp[i];

    const float2* si = (const float2*)&stageInv[wave][0];
    float2* gi = (float2*)(outInv + ((size_t)n * NPTS + p0) * NV);
    gi[lane] = si[lane];                           // 64 floats in one b64 round
  }
}

extern "C" void kernel_launch(void* const* d_in, const int* in_sizes, int n_in,
                              void* d_out, int out_size, void* d_ws, size_t ws_size,
                              hipStream_t stream) {
  (void)in_sizes; (void)n_in; (void)d_ws; (void)ws_size; (void)out_size;
  const float* xyz    = (const float*)d_in[0];
  const float* fm     = (const float*)d_in[1];
  const float* poses  = (const float*)d_in[2];
  const float* Ks     = (const float*)d_in[3];
  const float* emptyf = (const float*)d_in[4];

  float* outFeat = (float*)d_out;
  float* outInv  = outFeat + (size_t)N_B * NPTS * NV * FEAT_DIM;  // 82,400,000 floats

  dim3 grid((NPTS + WAVES * PTSW - 1) / (WAVES * PTSW), N_B);     // (782, 4)
  mvbts_kernel<<<grid, 128, 0, stream>>>(xyz, fm, poses, Ks, emptyf, outFeat, outInv);
}